// LightGCN_89670327206250
// MI455X (gfx1250) — compile-verified
//
#include <hip/hip_runtime.h>
#include <stdint.h>

#define EMB 64
#define WPB 8                 // waves per block (256 threads, wave32)
#define SPMM_BLOCKS 2048

#if defined(__has_builtin)
#if __has_builtin(__builtin_amdgcn_global_load_async_to_lds_b32)
#define HAVE_ASYNC_LDS 1
#endif
#endif
#ifndef HAVE_ASYNC_LDS
#define HAVE_ASYNC_LDS 0
#endif

#if defined(__has_builtin) && __has_builtin(__builtin_amdgcn_s_wait_asynccnt)
#define WAIT_ASYNC(n) __builtin_amdgcn_s_wait_asynccnt(n)
#else
#define WAIT_ASYNC(n) asm volatile("s_wait_asynccnt %0" ::"n"(n) : "memory")
#endif

// Guaranteed-native, non-returning, device-scope f32 atomic add.
// Non-returning => tracked on STOREcnt (cheap path); SCOPE_DEV is required
// because waves on different WGPs accumulate into the same rows.
__device__ __forceinline__ void gadd_f32(float* p, float v) {
  asm volatile("global_atomic_add_f32 %0, %1, off scope:SCOPE_DEV"
               :: "v"(p), "v"(v)
               : "memory");
}

// ---------------- SpMM scatter kernel ----------------
// One wave per 32-edge tile (grid-stride). Lanes split the 64-dim row: 2 dims
// per lane as float2. Edge indices are double-buffered into LDS with async
// global->LDS copies; each wave only touches LDS it staged itself, so the
// only sync needed is the wave's own s_wait_asynccnt.

__global__ __launch_bounds__(256) void k_spmm(
    const int* __restrict__ src, const int* __restrict__ dst,
    const float* __restrict__ dis,
    const float* __restrict__ h_in, float* __restrict__ h_out, int E) {
  __shared__ int s_src[WPB][2][32];
  __shared__ int s_dst[WPB][2][32];

  const int lane = threadIdx.x & 31;
  const int wv = threadIdx.x >> 5;
  const int waveId = blockIdx.x * WPB + wv;
  const int nWaves = gridDim.x * WPB;
  const int nTiles = (E + 31) >> 5;

  int t = waveId;
  int buf = 0;

  // per-edge body: all 32 lanes cooperate on one edge (2 dims/lane)
  auto edge = [&](int s, int d, int dpf) {
    if (dpf >= 0)
      __builtin_prefetch(h_in + (size_t)dpf * EMB, 0, 1);
    const float w = dis[s] * dis[d];
    const float2 hv = *((const float2*)(h_in + (size_t)d * EMB) + lane);
    float* op = h_out + (size_t)s * EMB + lane * 2;
    gadd_f32(op + 0, w * hv.x);
    gadd_f32(op + 1, w * hv.y);
  };

#if HAVE_ASYNC_LDS
  if (t < nTiles) {
    int e = (t << 5) + lane;
    if (e < E) {
      __builtin_amdgcn_global_load_async_to_lds_b32(
          const_cast<int*>(src + e), &s_src[wv][0][lane], 0, 0);
      __builtin_amdgcn_global_load_async_to_lds_b32(
          const_cast<int*>(dst + e), &s_dst[wv][0][lane], 0, 0);
    }
  }
#endif

  for (; t < nTiles; t += nWaves) {
#if HAVE_ASYNC_LDS
    int tn = t + nWaves;
    if (tn < nTiles) {
      // stage next tile into the other buffer, then wait for current tile
      // (async loads complete in order: waiting for <=2 outstanding means the
      // first two -- this tile's src/dst -- have landed in LDS).
      int e = (tn << 5) + lane;
      if (e < E) {
        __builtin_amdgcn_global_load_async_to_lds_b32(
            const_cast<int*>(src + e), &s_src[wv][buf ^ 1][lane], 0, 0);
        __builtin_amdgcn_global_load_async_to_lds_b32(
            const_cast<int*>(dst + e), &s_dst[wv][buf ^ 1][lane], 0, 0);
      }
      WAIT_ASYNC(2);
    } else {
      WAIT_ASYNC(0);
    }
#endif
    const int base = t << 5;
    if (base + 32 <= E) {
      // full tile: constant trip count, unrolled for MLP
#pragma unroll 4
      for (int i = 0; i < 32; ++i) {
#if HAVE_ASYNC_LDS
        edge(s_src[wv][buf][i], s_dst[wv][buf][i],
             (i + 6 < 32) ? s_dst[wv][buf][i + 6] : -1);
#else
        edge(src[base + i], dst[base + i],
             (i + 6 < 32) ? dst[base + i + 6] : -1);
#endif
      }
    } else {
      const int cnt = E - base;
      for (int i = 0; i < cnt; ++i) {
#if HAVE_ASYNC_LDS
        edge(s_src[wv][buf][i], s_dst[wv][buf][i], -1);
#else
        edge(src[base + i], dst[base + i], -1);
#endif
      }
    }
    buf ^= 1;
  }
}

// ---------------- elementwise kernels ----------------

__global__ __launch_bounds__(256) void k_init_x(
    const float* __restrict__ uw, const float* __restrict__ iw,
    float* __restrict__ h, float* __restrict__ acc, int nu_elems, int total_elems) {
  int i = (blockIdx.x * blockDim.x + threadIdx.x) * 4;
  if (i >= total_elems) return;
  float4 v = (i < nu_elems) ? *(const float4*)(uw + i)
                            : *(const float4*)(iw + (i - nu_elems));
  *(float4*)(h + i) = v;
  *(float4*)(acc + i) = v;
}

__global__ __launch_bounds__(256) void k_degree(
    const int* __restrict__ src, float* __restrict__ deg, int E) {
  int e = blockIdx.x * blockDim.x + threadIdx.x;
  if (e < E) gadd_f32(&deg[src[e]], 1.0f);
}

__global__ __launch_bounds__(256) void k_rsqrt(
    const float* __restrict__ deg, float* __restrict__ dis, int n) {
  int i = blockIdx.x * blockDim.x + threadIdx.x;
  if (i < n) dis[i] = rsqrtf(deg[i] + 1e-8f);
}

__global__ __launch_bounds__(256) void k_acc(
    float* __restrict__ acc, const float* __restrict__ h, float scale, int total_elems) {
  int i = (blockIdx.x * blockDim.x + threadIdx.x) * 4;
  if (i >= total_elems) return;
  float4 a = *(float4*)(acc + i);
  float4 b = *(const float4*)(h + i);
  a.x = (a.x + b.x) * scale;
  a.y = (a.y + b.y) * scale;
  a.z = (a.z + b.z) * scale;
  a.w = (a.w + b.w) * scale;
  *(float4*)(acc + i) = a;
}

// ---------------- launch ----------------

extern "C" void kernel_launch(void* const* d_in, const int* in_sizes, int n_in,
                              void* d_out, int out_size, void* d_ws, size_t ws_size,
                              hipStream_t stream) {
  const float* uw = (const float*)d_in[0];
  const float* iw = (const float*)d_in[1];
  const int* src = (const int*)d_in[2];
  const int* dst = (const int*)d_in[3];
  const int nu = in_sizes[0] / EMB;
  const int ni = in_sizes[1] / EMB;
  const int E = in_sizes[2];
  const int NT = nu + ni;
  const size_t NT64 = (size_t)NT * EMB;
  float* out = (float*)d_out;

  // workspace layout: h_a | h_b | deg | dis
  float* h_a = (float*)d_ws;
  float* h_b = h_a + NT64;
  float* deg = h_b + NT64;
  float* dis = deg + NT;

  (void)hipMemsetAsync(deg, 0, (size_t)NT * sizeof(float), stream);
  k_degree<<<(E + 255) / 256, 256, 0, stream>>>(src, deg, E);
  k_rsqrt<<<(NT + 255) / 256, 256, 0, stream>>>(deg, dis, NT);

  const int vec_blocks = (int)((NT64 / 4 + 255) / 256);
  k_init_x<<<vec_blocks, 256, 0, stream>>>(uw, iw, h_a, out, in_sizes[0], (int)NT64);

  float* hin = h_a;
  float* hout = h_b;
  for (int L = 0; L < 3; ++L) {
    (void)hipMemsetAsync(hout, 0, NT64 * sizeof(float), stream);
    k_spmm<<<SPMM_BLOCKS, 256, 0, stream>>>(src, dst, dis, hin, hout, E);
    const float scale = (L == 2) ? 0.25f : 1.0f;
    k_acc<<<vec_blocks, 256, 0, stream>>>(out, hout, scale, (int)NT64);
    float* tmp = hin; hin = hout; hout = tmp;
  }
}